// SpikeFP32ExtractLow5_43860206027307
// MI455X (gfx1250) — compile-verified
//
#include <hip/hip_runtime.h>
#include <stdint.h>

// ---------------------------------------------------------------------------
// SpikeFP32ExtractLow5 — memory-bound bit-extraction kernel for MI455X.
//
// Per element (32 floats of 0.0/1.0 = bits of an fp32):
//   e    = bits[1..8]  (8-bit exponent, MSB first)
//   mant = bits[9..12] (top 4 mantissa bits)
//   s    = e - 127
//   out5 = (0 <= s <= 4) ? ((16 | mant) >> (4 - s)) : 0
//   out[j] = bit (4-j) of out5,  j = 0..4  (written as 0.0f / 1.0f)
//
// Traffic: read 52 of 128 B/elem (=> one 64B HBM sector), write 20 B/elem
//   => ~176 MB @ 23.3 TB/s ~= 7.6 us. Pure streaming; no matmul structure,
//   so the CDNA5 paths used are the async global->LDS data mover (ASYNCcnt,
//   s_wait_asynccnt) and non-temporal cache hints on both loads and stores
//   (268 MB stream/pass > 192 MB L2 — RT policy would only thrash).
// ---------------------------------------------------------------------------

typedef __attribute__((ext_vector_type(4))) float f4;

__device__ __forceinline__ uint32_t bitof(float f) { return f != 0.0f ? 1u : 0u; }

__device__ __forceinline__ uint32_t pack5(uint32_t e, uint32_t mant) {
  uint32_t s  = e - 127u;                      // unsigned wrap for e < 127
  uint32_t sh = (s <= 4u) ? (4u - s) : 0u;     // avoid UB shift amounts
  uint32_t v  = (16u | mant) >> sh;
  return (s <= 4u) ? v : 0u;
}

__device__ __forceinline__ void store5(float* __restrict__ o, uint32_t out5) {
  // out = [bit4, bit3, bit2, bit1, bit0] as floats; streamed (non-temporal).
  __builtin_nontemporal_store((float)((out5 >> 4) & 1u), o + 0);
  __builtin_nontemporal_store((float)((out5 >> 3) & 1u), o + 1);
  __builtin_nontemporal_store((float)((out5 >> 2) & 1u), o + 2);
  __builtin_nontemporal_store((float)((out5 >> 1) & 1u), o + 3);
  __builtin_nontemporal_store((float)( out5       & 1u), o + 4);
}

// --- async global -> LDS issue: 52 bytes (floats 0..12) of one element -----
// The immediate offset is added to both the LDS and the global address
//      -> one offset serves both sides.
// th:TH_LOAD_NT — single-use streaming data, don't retain in L2.
__device__ __forceinline__ void issue_elem_async(const float* gp, uint32_t lds_byte) {
  uint64_t g = (uint64_t)(uintptr_t)gp;
  asm volatile("global_load_async_to_lds_b128 %0, %1, off th:TH_LOAD_NT"
               :: "v"(lds_byte), "v"(g) : "memory");
  asm volatile("global_load_async_to_lds_b128 %0, %1, off offset:16 th:TH_LOAD_NT"
               :: "v"(lds_byte), "v"(g) : "memory");
  asm volatile("global_load_async_to_lds_b128 %0, %1, off offset:32 th:TH_LOAD_NT"
               :: "v"(lds_byte), "v"(g) : "memory");
  asm volatile("global_load_async_to_lds_b32 %0, %1, off offset:48 th:TH_LOAD_NT"
               :: "v"(lds_byte), "v"(g) : "memory");
}

// 13 dwords per element in LDS (floats 0..12). Odd stride => lane L reads at
// dword 13*L + k, 13 invertible mod 64 => bank-conflict-free readback (the
// compiler merges the 12 dword reads into 6 ds_load_2addr_b32, still
// conflict-free per address).
#define ELEM_DW 13

__global__ void __launch_bounds__(256)
spike_extract_async(const float* __restrict__ x, float* __restrict__ out, int iters) {
  __shared__ float lds[2 * 256 * ELEM_DW];
  const int tid = threadIdx.x;
  const long long stride = (long long)gridDim.x * 256;  // elements per iteration
  long long elem = (long long)blockIdx.x * 256 + tid;

  // LDS byte offsets of this thread's two buffers. Low 32 bits of the generic
  // pointer to a __shared__ object are the LDS byte address (flat aperture);
  // confirmed by round-1 codegen (v1 = 13*tid*4).
  uint32_t lds_off0 = (uint32_t)(uintptr_t)(&lds[tid * ELEM_DW]);
  uint32_t lds_off1 = (uint32_t)(uintptr_t)(&lds[(256 + tid) * ELEM_DW]);

  if (iters <= 0) return;

  // Prologue: prefetch iteration 0 into buffer 0. 4 async ops / wave / tile.
  issue_elem_async(x + elem * 32, lds_off0);

  int buf = 0;
  for (int i = 0; i < iters; ++i) {
    long long next = elem + stride;
    if (i + 1 < iters) {
      // Double buffer: start next tile (outstanding -> 8), then wait until
      // only the next tile's 4 remain => current tile's data is in LDS.
      // Async loads complete in order within a wave, so cnt<=4 is exact.
      issue_elem_async(x + next * 32, (buf == 0) ? lds_off1 : lds_off0);
      asm volatile("s_wait_asynccnt 4" ::: "memory");
    } else {
      asm volatile("s_wait_asynccnt 0" ::: "memory");
    }

    const float* w = &lds[(buf ? 256 * ELEM_DW : 0) + tid * ELEM_DW];
    uint32_t e = (bitof(w[1]) << 7) | (bitof(w[2]) << 6) | (bitof(w[3]) << 5) |
                 (bitof(w[4]) << 4) | (bitof(w[5]) << 3) | (bitof(w[6]) << 2) |
                 (bitof(w[7]) << 1) |  bitof(w[8]);
    uint32_t mant = (bitof(w[9]) << 3) | (bitof(w[10]) << 2) |
                    (bitof(w[11]) << 1) | bitof(w[12]);
    store5(out + elem * 5, pack5(e, mant));

    elem = next;
    buf ^= 1;
  }
}

// --- fallback: direct vectorized loads (used only if sizes don't divide) ---
__global__ void __launch_bounds__(256)
spike_extract_direct(const float* __restrict__ x, float* __restrict__ out, long long n) {
  long long i = (long long)blockIdx.x * blockDim.x + threadIdx.x;
  if (i >= n) return;
  const float* p = x + i * 32;
  f4 a = __builtin_nontemporal_load((const f4*)p);        // bits 0..3
  f4 b = __builtin_nontemporal_load((const f4*)(p + 4));  // bits 4..7
  f4 c = __builtin_nontemporal_load((const f4*)(p + 8));  // bits 8..11
  float d = __builtin_nontemporal_load(p + 12);           // bit 12 (m3)

  uint32_t e = (bitof(a.y) << 7) | (bitof(a.z) << 6) | (bitof(a.w) << 5) |
               (bitof(b.x) << 4) | (bitof(b.y) << 3) | (bitof(b.z) << 2) |
               (bitof(b.w) << 1) |  bitof(c.x);
  uint32_t mant = (bitof(c.y) << 3) | (bitof(c.z) << 2) |
                  (bitof(c.w) << 1) | bitof(d);
  store5(out + i * 5, pack5(e, mant));
}

extern "C" void kernel_launch(void* const* d_in, const int* in_sizes, int n_in,
                              void* d_out, int out_size, void* d_ws, size_t ws_size,
                              hipStream_t stream) {
  (void)n_in; (void)out_size; (void)d_ws; (void)ws_size;
  const float* x = (const float*)d_in[0];
  float* out = (float*)d_out;
  const long long nelem = (long long)in_sizes[0] / 32;   // 2048*1024 = 2,097,152

  const int  T   = 256;
  const long long BLK = 2048;                            // persistent-style grid
  if (nelem > 0 && (nelem % (BLK * T)) == 0) {
    const int iters = (int)(nelem / (BLK * T));          // 4 for the ref shape
    spike_extract_async<<<dim3((uint32_t)BLK), dim3(T), 0, stream>>>(x, out, iters);
  } else {
    const long long nb = (nelem + T - 1) / T;
    spike_extract_direct<<<dim3((uint32_t)nb), dim3(T), 0, stream>>>(x, out, nelem);
  }
}